// AffineLabelAttention_56049323213774
// MI455X (gfx1250) — compile-verified
//
#include <hip/hip_runtime.h>
#include <hip/hip_bf16.h>

typedef __attribute__((ext_vector_type(2))) float v2f;
typedef __attribute__((ext_vector_type(8))) float v8f;

#define DIM      512
#define NLAB     64
#define BATCH    8
#define SEQ      512
#define ROWS     (BATCH * SEQ)          // 4096  (M dimension)
#define NROWS_HD (BATCH * NLAB * SEQ)   // 262144 rows of hb/db

// ---------------------------------------------------------------------------
// Kernel 1: f32 WMMA GEMM.
//   hb[(b*64+l)*512 + i] = sum_d head[b,i,d] * label_W[l, d]        + bias[l]
//   db[(b*64+l)*512 + j] = sum_d dep [b,j,d] * label_W[l, 512 + d]
// One wave = one 16x16 output tile.  M=4096, N=64, K=512 -> 256*4 = 1024
// tiles per matrix, 2048 waves total.
//
// V_WMMA_F32_16X16X4_F32 operand layout (wave32):
//   A (16x4):  lane L<16  -> v0=A[M=L][k0+0], v1=A[M=L][k0+1]
//              lane L>=16 -> v0=A[M=L-16][k0+2], v1=A[M=L-16][k0+3]
//   B (4x16):  lane L<16  -> v0=B[k0+0][N=L], v1=B[k0+1][N=L]
//              lane L>=16 -> v0=B[k0+2][N=L-16], v1=B[k0+3][N=L-16]
//   -> per lane, per K-step: one contiguous float2 load from X and from W.
//   C/D (16x16): VGPR r: lanes 0-15 = row M=r, lanes 16-31 = row M=8+r.
// ---------------------------------------------------------------------------
__global__ __launch_bounds__(256)
void affine_gemm_wmma(const float* __restrict__ head,
                      const float* __restrict__ dep,
                      const float* __restrict__ label_W,
                      const float* __restrict__ label_b,
                      float* __restrict__ hb,
                      float* __restrict__ db)
{
    const int wavesPerBlock = blockDim.x >> 5;
    const int waveId = blockIdx.x * wavesPerBlock + (threadIdx.x >> 5);
    const int lane   = threadIdx.x & 31;

    const bool isDep = waveId >= 1024;
    const int  tile  = isDep ? (waveId - 1024) : waveId;
    const int  n0    = (tile & 3) << 4;   // label tile  (4 tiles of 16)
    const int  m0    = (tile >> 2) << 4;  // row tile    (256 tiles of 16)

    const float* __restrict__ X  = isDep ? dep : head;
    const int wOff  = isDep ? DIM : 0;

    const int lsub  = lane & 15;
    const int khalf = (lane >> 4) << 1;   // 0 for lanes 0-15, 2 for 16-31

    const float* aptr = X       + (size_t)(m0 + lsub) * DIM       + khalf;
    const float* bptr = label_W + (size_t)(n0 + lsub) * (2 * DIM) + wOff + khalf;

    v8f acc = {};
#pragma unroll 8
    for (int k = 0; k < DIM; k += 4) {
        v2f a = *(const v2f*)(aptr + k);
        v2f b = *(const v2f*)(bptr + k);
        acc = __builtin_amdgcn_wmma_f32_16x16x4_f32(
            /*neg_a=*/false, a, /*neg_b=*/false, b,
            /*c_mod=*/(short)0, acc, /*reuse_a=*/false, /*reuse_b=*/false);
    }

    // Epilogue: scatter the 16x16 tile into [b][l][i] order (+bias for head).
    const int   l    = n0 + lsub;
    const float bias = isDep ? 0.0f : label_b[l];
    float* __restrict__ OUT = isDep ? db : hb;
    const int mbase = m0 + ((lane >> 4) << 3);   // +8 for upper half-wave

#pragma unroll
    for (int r = 0; r < 8; ++r) {
        const int M  = mbase + r;          // global row = b*512 + i
        const int bi = M >> 9;             // batch
        const int i  = M & 511;            // position
        OUT[(((size_t)(bi * NLAB + l)) << 9) + i] = acc[r] + bias;
    }
}

// ---------------------------------------------------------------------------
// Kernel 2: broadcast add (bandwidth-bound, 512 MiB of stores).
//   out[row*512 + j] = hb[row] + db[(row>>9)*512 + j]
// One 128-thread block per row; float4 (b128) traffic.  db rows are reused
// across all 512 values of i per (b,l) -> served from L2.
// ---------------------------------------------------------------------------
__global__ __launch_bounds__(128)
void affine_bcast_add(const float* __restrict__ hb,
                      const float* __restrict__ db,
                      float4* __restrict__ out)
{
    const int row  = blockIdx.x;                 // (b*64 + l)*512 + i
    const float s  = hb[row];
    const int lrow = row >> 9;                   // b*64 + l

    const float4* __restrict__ d4 =
        (const float4*)(db + ((size_t)lrow << 9));

    const float4 v = d4[threadIdx.x];
    float4 o;
    o.x = s + v.x;
    o.y = s + v.y;
    o.z = s + v.z;
    o.w = s + v.w;
    out[((size_t)row << 7) + threadIdx.x] = o;
}

extern "C" void kernel_launch(void* const* d_in, const int* in_sizes, int n_in,
                              void* d_out, int out_size, void* d_ws, size_t ws_size,
                              hipStream_t stream) {
    const float* head    = (const float*)d_in[0];   // (8, 512, 512)
    const float* dep     = (const float*)d_in[1];   // (8, 512, 512)
    const float* label_W = (const float*)d_in[2];   // (64, 1024)
    const float* label_b = (const float*)d_in[3];   // (64,)

    float* hb = (float*)d_ws;                       // 262144 floats (1 MiB)
    float* db = hb + NROWS_HD;                      // 262144 floats (1 MiB)

    // 2048 tile-waves, 8 waves (256 threads) per block -> 256 blocks.
    affine_gemm_wmma<<<256, 256, 0, stream>>>(head, dep, label_W, label_b, hb, db);

    // 262144 rows, one block per row.
    affine_bcast_add<<<NROWS_HD, 128, 0, stream>>>(hb, db, (float4*)d_out);
}